// Elastic_60095182405994
// MI455X (gfx1250) — compile-verified
//
#include <hip/hip_runtime.h>

// ---------------- problem constants (match reference) ----------------
constexpr int   GN      = 512;          // NY == NX
constexpr int   NSHOT   = 4;
constexpr int   NSRC    = 2;
constexpr int   NREC    = 64;
constexpr int   NT      = 32;
constexpr float DT_     = 0.0005f;
constexpr float INV_DX  = 1.0f / 5.0f;
constexpr float C1      = 9.0f / 8.0f;
constexpr float C2      = -1.0f / 24.0f;

constexpr int TILE = 32;                // computed tile (32x32 per block)
constexpr int TL   = TILE + 4;          // staged tile with +/-2 halo = 36
#define L(a, b) ((a) * TL + (b))

// field slots inside workspace: 0 vy, 1 vx, 2 syy, 3 sxy, 4 sxx,
// 5 m_vyy, 6 m_vyx, 7 m_vxy, 8 m_vxx, 9 m_syyy, 10 m_sxyy, 11 m_sxyx, 12 m_sxxx
constexpr size_t FS = (size_t)NSHOT * GN * GN;   // elements per field

// ---------------- CDNA5 Tensor Data Mover staging ----------------
#if defined(__has_builtin)
#  if __has_builtin(__builtin_amdgcn_tensor_load_to_lds)
#    define HAVE_TDM 1
#  endif
#endif
#ifndef HAVE_TDM
#  define HAVE_TDM 0
#endif

#if HAVE_TDM
typedef __attribute__((ext_vector_type(4))) unsigned int tdm_v4u;
typedef __attribute__((ext_vector_type(8))) int          tdm_v8i;
typedef __attribute__((ext_vector_type(4))) int          tdm_v4i;

// DMA a 2-D tile (tile_w x tile_h f32, row stride g_stride elems) from global
// into LDS (packed rows of tile_w). rem_w/rem_h = remaining tensor extent from
// the tile origin (far-edge OOB returns zero per TDM spec; callers guarantee
// the near edge is in bounds).
__device__ __forceinline__ void tdm_load_tile(float* lds_tile, const float* gsrc,
                                              int tile_w, int tile_h,
                                              int g_stride, int rem_w, int rem_h) {
  unsigned long long ga = (unsigned long long)(__SIZE_TYPE__)gsrc;
  unsigned int lds = (unsigned int)(__SIZE_TYPE__)lds_tile; // low 32b of flat = LDS offset
  tdm_v4u g0;
  g0[0] = 1u;                                                   // count=1, user descriptor
  g0[1] = lds;                                                  // lds_addr
  g0[2] = (unsigned int)(ga & 0xFFFFFFFFull);                   // global_addr[31:0]
  g0[3] = ((unsigned int)(ga >> 32) & 0x01FFFFFFu) | 0x80000000u; // addr[56:32] | type=2
  tdm_v8i g1;
  g1[0] = (int)(2u << 16);                                      // data_size = 4B
  g1[1] = (int)(((unsigned)rem_w & 0xFFFFu) << 16);             // tensor_dim0[15:0]
  g1[2] = (int)((((unsigned)rem_w >> 16) & 0xFFFFu) |
                (((unsigned)rem_h & 0xFFFFu) << 16));           // dim0 hi | dim1 lo
  g1[3] = (int)((((unsigned)rem_h >> 16) & 0xFFFFu) |
                ((unsigned)tile_w << 16));                      // dim1 hi | tile_dim0
  g1[4] = tile_h & 0xFFFF;                                      // tile_dim1 (tile_dim2=0)
  g1[5] = g_stride;                                             // tensor_dim0_stride lo32
  g1[6] = 0;
  g1[7] = 0;
  tdm_v4i z4 = {0, 0, 0, 0};
#if defined(__clang_major__) && __clang_major__ >= 23
  tdm_v8i z8 = {0, 0, 0, 0, 0, 0, 0, 0};
  __builtin_amdgcn_tensor_load_to_lds(g0, g1, z4, z4, z8, 0);
#else
  __builtin_amdgcn_tensor_load_to_lds(g0, g1, z4, z4, 0);
#endif
}

__device__ __forceinline__ void tdm_wait_all() {
#if __has_builtin(__builtin_amdgcn_s_wait_tensorcnt)
  __builtin_amdgcn_s_wait_tensorcnt(0);
#else
  asm volatile("s_wait_tensorcnt 0x0" ::: "memory");
#endif
}
#endif // HAVE_TDM

// zero-padded cooperative fallback (boundary tiles)
__device__ __forceinline__ void coop_load_tile(float* tile, const float* __restrict__ g,
                                               int y0, int x0, int tid) {
  for (int i = tid; i < TL * TL; i += 256) {
    int ry = i / TL, rx = i - ry * TL;
    int gy = y0 - 2 + ry, gx = x0 - 2 + rx;
    float v = 0.0f;
    if (gy >= 0 && gy < GN && gx >= 0 && gx < GN) v = g[gy * GN + gx];
    tile[i] = v;
  }
}

// ---------------- init kernels ----------------
__global__ void zero_kernel(float4* __restrict__ p, long long n4) {
  long long i = (long long)blockIdx.x * blockDim.x + threadIdx.x;
  long long stride = (long long)gridDim.x * blockDim.x;
  float4 z = make_float4(0.f, 0.f, 0.f, 0.f);
  for (; i < n4; i += stride) p[i] = z;
}

__global__ void profile_kernel(float* __restrict__ aprof, float* __restrict__ bprof) {
  int i = blockIdx.x * blockDim.x + threadIdx.x;
  if (i >= GN) return;
  const float w = 20.0f;                          // PML_WIDTH
  float fi = (float)i;
  float frac = fmaxf((w - fi) / w, (fi - (float)(GN - 1 - 20)) / w);
  frac = fminf(fmaxf(frac, 0.0f), 1.0f);
  float sigma = 3.0f * 3500.0f * 6.907755278982137f / (2.0f * 20.0f * 5.0f) * frac * frac;
  float alpha = 78.53981633974483f;               // pi * 25
  float b = expf(-(sigma + alpha) * DT_);
  float a = sigma / (sigma + alpha) * (b - 1.0f);
  aprof[i] = a;
  bprof[i] = b;
}

// ---------------- velocity + PML-mem update (+ fused source injection) ----------------
__global__ __launch_bounds__(256) void vel_step_kernel(
    float* __restrict__ ws, const float* __restrict__ buoy,
    const float* __restrict__ amps, const int* __restrict__ sloc, int step) {
  __shared__ float t_syy[TL * TL];
  __shared__ float t_sxy[TL * TL];
  __shared__ float t_sxx[TL * TL];

  const int shot = blockIdx.z;
  const int x0 = blockIdx.x * TILE, y0 = blockIdx.y * TILE;
  const size_t so = (size_t)shot * GN * GN;

  float*       vy     = ws + 0 * FS + so;
  float*       vx     = ws + 1 * FS + so;
  const float* syy    = ws + 2 * FS + so;
  const float* sxy    = ws + 3 * FS + so;
  const float* sxx    = ws + 4 * FS + so;
  float*       m_syyy = ws + 9 * FS + so;
  float*       m_sxyy = ws + 10 * FS + so;
  float*       m_sxyx = ws + 11 * FS + so;
  float*       m_sxxx = ws + 12 * FS + so;
  const float* aprof  = ws + 13 * FS;
  const float* bprof  = aprof + GN;

  const int tid = threadIdx.y * 32 + threadIdx.x;

#if HAVE_TDM
  const bool interior = (x0 >= 2) && (y0 >= 2) &&
                        (x0 + TILE + 2 <= GN) && (y0 + TILE + 2 <= GN);
  if (interior) {
    if (threadIdx.y == 0) {                        // one wave drives the TDM
      const size_t org = (size_t)(y0 - 2) * GN + (x0 - 2);
      tdm_load_tile(t_syy, syy + org, TL, TL, GN, GN - (x0 - 2), GN - (y0 - 2));
      tdm_load_tile(t_sxy, sxy + org, TL, TL, GN, GN - (x0 - 2), GN - (y0 - 2));
      tdm_load_tile(t_sxx, sxx + org, TL, TL, GN, GN - (x0 - 2), GN - (y0 - 2));
      tdm_wait_all();
    }
  } else
#endif
  {
    coop_load_tile(t_syy, syy, y0, x0, tid);
    coop_load_tile(t_sxy, sxy, y0, x0, tid);
    coop_load_tile(t_sxx, sxx, y0, x0, tid);
  }
  __syncthreads();

#pragma unroll
  for (int k = 0; k < 4; ++k) {
    const int y = y0 + threadIdx.y + k * 8;
    const int x = x0 + threadIdx.x;
    const int ly = threadIdx.y + k * 8 + 2;
    const int lx = threadIdx.x + 2;
    const int idx = y * GN + x;

    const float ay = aprof[y], by = bprof[y];
    const float ax = aprof[x], bx = bprof[x];

    // dsyy = diff_minus(syy, y)
    float dsyy = (C1 * (t_syy[L(ly, lx)] - t_syy[L(ly - 1, lx)]) +
                  C2 * (t_syy[L(ly + 1, lx)] - t_syy[L(ly - 2, lx)])) * INV_DX;
    float msyyy = by * m_syyy[idx] + ay * dsyy;
    m_syyy[idx] = msyyy;

    // dsxy_x = diff_plus(sxy, x)
    float dsxyx = (C1 * (t_sxy[L(ly, lx + 1)] - t_sxy[L(ly, lx)]) +
                   C2 * (t_sxy[L(ly, lx + 2)] - t_sxy[L(ly, lx - 1)])) * INV_DX;
    float msxyx = bx * m_sxyx[idx] + ax * dsxyx;
    m_sxyx[idx] = msxyx;

    const float bu = buoy[idx];
    float v = vy[idx] + DT_ * bu * ((dsyy + msyyy) + (dsxyx + msxyx));

    // fused point-source injection (scalar loads; indices uniform per shot)
#pragma unroll
    for (int s = 0; s < NSRC; ++s) {
      const int sy = sloc[(shot * NSRC + s) * 2 + 0];
      const int sx = sloc[(shot * NSRC + s) * 2 + 1];
      if (y == sy && x == sx)
        v += amps[(shot * NSRC + s) * NT + step] * DT_ * bu;
    }
    vy[idx] = v;

    // dsxy_y = diff_plus(sxy, y)
    float dsxyy = (C1 * (t_sxy[L(ly + 1, lx)] - t_sxy[L(ly, lx)]) +
                   C2 * (t_sxy[L(ly + 2, lx)] - t_sxy[L(ly - 1, lx)])) * INV_DX;
    float msxyy = by * m_sxyy[idx] + ay * dsxyy;
    m_sxyy[idx] = msxyy;

    // dsxx = diff_minus(sxx, x)
    float dsxx = (C1 * (t_sxx[L(ly, lx)] - t_sxx[L(ly, lx - 1)]) +
                  C2 * (t_sxx[L(ly, lx + 1)] - t_sxx[L(ly, lx - 2)])) * INV_DX;
    float msxxx = bx * m_sxxx[idx] + ax * dsxx;
    m_sxxx[idx] = msxxx;

    vx[idx] = vx[idx] + DT_ * bu * ((dsxyy + msxyy) + (dsxx + msxxx));
  }
}

// ---------------- stress + PML-mem update ----------------
__global__ __launch_bounds__(256) void stress_step_kernel(
    float* __restrict__ ws, const float* __restrict__ lamb,
    const float* __restrict__ mu) {
  __shared__ float t_vy[TL * TL];
  __shared__ float t_vx[TL * TL];

  const int shot = blockIdx.z;
  const int x0 = blockIdx.x * TILE, y0 = blockIdx.y * TILE;
  const size_t so = (size_t)shot * GN * GN;

  const float* vy    = ws + 0 * FS + so;
  const float* vx    = ws + 1 * FS + so;
  float*       syy   = ws + 2 * FS + so;
  float*       sxy   = ws + 3 * FS + so;
  float*       sxx   = ws + 4 * FS + so;
  float*       m_vyy = ws + 5 * FS + so;
  float*       m_vyx = ws + 6 * FS + so;
  float*       m_vxy = ws + 7 * FS + so;
  float*       m_vxx = ws + 8 * FS + so;
  const float* aprof = ws + 13 * FS;
  const float* bprof = aprof + GN;

  const int tid = threadIdx.y * 32 + threadIdx.x;

#if HAVE_TDM
  const bool interior = (x0 >= 2) && (y0 >= 2) &&
                        (x0 + TILE + 2 <= GN) && (y0 + TILE + 2 <= GN);
  if (interior) {
    if (threadIdx.y == 0) {
      const size_t org = (size_t)(y0 - 2) * GN + (x0 - 2);
      tdm_load_tile(t_vy, vy + org, TL, TL, GN, GN - (x0 - 2), GN - (y0 - 2));
      tdm_load_tile(t_vx, vx + org, TL, TL, GN, GN - (x0 - 2), GN - (y0 - 2));
      tdm_wait_all();
    }
  } else
#endif
  {
    coop_load_tile(t_vy, vy, y0, x0, tid);
    coop_load_tile(t_vx, vx, y0, x0, tid);
  }
  __syncthreads();

#pragma unroll
  for (int k = 0; k < 4; ++k) {
    const int y = y0 + threadIdx.y + k * 8;
    const int x = x0 + threadIdx.x;
    const int ly = threadIdx.y + k * 8 + 2;
    const int lx = threadIdx.x + 2;
    const int idx = y * GN + x;

    const float ay = aprof[y], by = bprof[y];
    const float ax = aprof[x], bx = bprof[x];

    // dvyy = diff_plus(vy, y)
    float dvyy = (C1 * (t_vy[L(ly + 1, lx)] - t_vy[L(ly, lx)]) +
                  C2 * (t_vy[L(ly + 2, lx)] - t_vy[L(ly - 1, lx)])) * INV_DX;
    float mvyy = by * m_vyy[idx] + ay * dvyy;
    m_vyy[idx] = mvyy;

    // dvxx = diff_plus(vx, x)
    float dvxx = (C1 * (t_vx[L(ly, lx + 1)] - t_vx[L(ly, lx)]) +
                  C2 * (t_vx[L(ly, lx + 2)] - t_vx[L(ly, lx - 1)])) * INV_DX;
    float mvxx = bx * m_vxx[idx] + ax * dvxx;
    m_vxx[idx] = mvxx;

    const float tvyy = dvyy + mvyy;
    const float tvxx = dvxx + mvxx;
    const float la = lamb[idx];
    const float muv = mu[idx];
    const float l2m = la + 2.0f * muv;

    syy[idx] = syy[idx] + DT_ * (l2m * tvyy + la * tvxx);
    sxx[idx] = sxx[idx] + DT_ * (l2m * tvxx + la * tvyy);

    // dvyx = diff_minus(vy, x)
    float dvyx = (C1 * (t_vy[L(ly, lx)] - t_vy[L(ly, lx - 1)]) +
                  C2 * (t_vy[L(ly, lx + 1)] - t_vy[L(ly, lx - 2)])) * INV_DX;
    float mvyx = bx * m_vyx[idx] + ax * dvyx;
    m_vyx[idx] = mvyx;

    // dvxy = diff_minus(vx, y)
    float dvxy = (C1 * (t_vx[L(ly, lx)] - t_vx[L(ly - 1, lx)]) +
                  C2 * (t_vx[L(ly + 1, lx)] - t_vx[L(ly - 2, lx)])) * INV_DX;
    float mvxy = by * m_vxy[idx] + ay * dvxy;
    m_vxy[idx] = mvxy;

    sxy[idx] = sxy[idx] + DT_ * muv * ((dvyx + mvyx) + (dvxy + mvxy));
  }
}

// ---------------- receiver sampling ----------------
__global__ void rec_kernel(const float* __restrict__ ws, const int* __restrict__ rloc,
                           float* __restrict__ out, int step) {
  int tid = blockIdx.x * blockDim.x + threadIdx.x;
  if (tid >= NSHOT * NREC) return;
  int shot = tid / NREC;
  int ry = rloc[tid * 2 + 0];
  int rx = rloc[tid * 2 + 1];
  out[tid * NT + step] = ws[0 * FS + (size_t)shot * GN * GN + (size_t)ry * GN + rx];
}

// ---------------- host launcher ----------------
extern "C" void kernel_launch(void* const* d_in, const int* in_sizes, int n_in,
                              void* d_out, int out_size, void* d_ws, size_t ws_size,
                              hipStream_t stream) {
  (void)in_sizes; (void)n_in; (void)out_size; (void)ws_size;
  const float* lamb = (const float*)d_in[0];
  const float* mu   = (const float*)d_in[1];
  const float* buoy = (const float*)d_in[2];
  const float* amps = (const float*)d_in[3];
  const int*   sloc = (const int*)d_in[4];
  const int*   rloc = (const int*)d_in[5];
  float*       out  = (float*)d_out;
  float*       ws   = (float*)d_ws;

  // zero the 13 state fields (ws is poisoned by the harness)
  const long long n4 = (long long)(13 * FS / 4);
  zero_kernel<<<2048, 256, 0, stream>>>((float4*)ws, n4);
  // PML profiles (ny == nx, one pair serves both axes)
  profile_kernel<<<2, 256, 0, stream>>>(ws + 13 * FS, ws + 13 * FS + GN);

  const dim3 grid(GN / TILE, GN / TILE, NSHOT);  // (16,16,4)
  const dim3 block(32, 8);
  for (int t = 0; t < NT; ++t) {
    vel_step_kernel<<<grid, block, 0, stream>>>(ws, buoy, amps, sloc, t);
    stress_step_kernel<<<grid, block, 0, stream>>>(ws, lamb, mu);
    rec_kernel<<<1, 256, 0, stream>>>(ws, rloc, out, t);
  }
}